// Len_AvgPool_77403900608620
// MI455X (gfx1250) — compile-verified
//
#include <hip/hip_runtime.h>

typedef __attribute__((ext_vector_type(2))) float v2f;
typedef __attribute__((ext_vector_type(8))) float v8f;

#define B_  64
#define T_  2048
#define D_  768
#define EPS_ 1e-5f

// ---------------------------------------------------------------------------
// Kernel 1: exact global min/max of gnoise_raw [B,D] -> mm[0]=min, mm[1]=max
// Single workgroup, fixed-order tree reduction (deterministic; fp min/max is
// order-independent anyway).
// ---------------------------------------------------------------------------
__global__ __launch_bounds__(1024) void gminmax_kernel(const float* __restrict__ g,
                                                       float* __restrict__ mm) {
    __shared__ float smn[1024];
    __shared__ float smx[1024];
    const int tid = threadIdx.x;
    float mn =  3.402823466e38f;
    float mx = -3.402823466e38f;
    for (int i = tid; i < B_ * D_; i += 1024) {
        float v = g[i];
        mn = fminf(mn, v);
        mx = fmaxf(mx, v);
    }
    smn[tid] = mn; smx[tid] = mx;
    __syncthreads();
    for (int s = 512; s > 0; s >>= 1) {
        if (tid < s) {
            smn[tid] = fminf(smn[tid], smn[tid + s]);
            smx[tid] = fmaxf(smx[tid], smx[tid + s]);
        }
        __syncthreads();
    }
    if (tid == 0) { mm[0] = smn[0]; mm[1] = smx[0]; }
}

// ---------------------------------------------------------------------------
// Kernel 2: masked + full partial sums over a t-chunk via V_WMMA_F32_16X16X4_F32.
//
//   D(16x16) += A(16x4) * B(4x16)
//   A row 0 = mask(t+k),  A row 1 = 1.0, rows 2..15 = 0
//   B[k][n] = outputs[b][t+k][d0+n]
//   => D[0][n] = masked partial sum of column d0+n, D[1][n] = full partial sum.
//
// f32 operand layout (per ISA 16x4 A table, mirrored for B): lanes 0-15 hold
// K={0,1} in the 2 VGPRs, lanes 16-31 hold K={2,3}. C/D: VGPR0 lanes0-15 = M=0,
// VGPR1 lanes0-15 = M=1, N = lane.
//
// Block = 256 thr (8 wave32); wave w owns d columns [w*96, w*96+96) as six
// 16-wide tiles -> six WMMA accumulators. Lanes 0-15 of a wave read 64
// contiguous bytes per row; the 8 waves + 6 tiles cover the full 3 KB row, so
// HBM traffic is one pass over `outputs` (the chunk stays L2-resident across
// the j tiles).
// ---------------------------------------------------------------------------
template <bool FUSED>
__global__ __launch_bounds__(256) void pool_kernel(const float* __restrict__ xs,
                                                   const float* __restrict__ lens,
                                                   const float* __restrict__ graw,
                                                   const float* __restrict__ mm,
                                                   float* __restrict__ wsm,
                                                   float* __restrict__ wsf,
                                                   float* __restrict__ outp,
                                                   int tchunks) {
    const int tc    = blockIdx.x;          // t-chunk
    const int b     = blockIdx.y;          // batch row
    const int TS    = T_ / tchunks;        // frames in this chunk (multiple of 4)
    const int lane  = threadIdx.x & 31;
    const int wave  = threadIdx.x >> 5;
    const int m     = lane & 15;           // M row (A) / N column (B,C,D)
    const int khalf = lane >> 4;           // 0: K in {0,1}; 1: K in {2,3}

    const float length = lens[b] * (float)T_;
    int count = (int)floorf(length) + 1;
    count = count < 1 ? 1 : (count > T_ ? T_ : count);

    // A-matrix row selectors (branch-free): row0 -> mask, row1 -> ones
    const float a_is0 = (m == 0) ? 1.0f : 0.0f;
    const float a_is1 = (m == 1) ? 1.0f : 0.0f;

    // column base for this lane: d = wave*96 + j*16 + m
    const float* base = xs + (size_t)b * T_ * D_ + (size_t)wave * 96 + m;

    v8f acc[6];
#pragma unroll
    for (int j = 0; j < 6; ++j) acc[j] = (v8f){0.f,0.f,0.f,0.f,0.f,0.f,0.f,0.f};

    const int t_begin = tc * TS;
    const int t_end   = t_begin + TS;
    for (int t0 = t_begin; t0 < t_end; t0 += 4) {
        const int k0 = t0 + 2 * khalf;                       // row for VGPR0
        const float mk0 = (k0     < count) ? 1.0f : 0.0f;
        const float mk1 = (k0 + 1 < count) ? 1.0f : 0.0f;
        v2f a;
        a.x = a_is0 * mk0 + a_is1;                           // A[m][k0]
        a.y = a_is0 * mk1 + a_is1;                           // A[m][k0+1]

        const float* prow0 = base + (size_t)k0 * D_;
        const float* prow1 = prow0 + D_;
#pragma unroll
        for (int j = 0; j < 6; ++j) {
            v2f bm;
            bm.x = prow0[j * 16];                            // B[k0  ][n]
            bm.y = prow1[j * 16];                            // B[k0+1][n]
            acc[j] = __builtin_amdgcn_wmma_f32_16x16x4_f32(
                false, a, false, bm, (short)0, acc[j], false, false);
        }
    }

    // D[0][n] (masked) lives in acc[j][0] lanes 0-15; D[1][n] (full) in acc[j][1].
    if (khalf == 0) {
#pragma unroll
        for (int j = 0; j < 6; ++j) {
            const int   d    = wave * 96 + j * 16 + m;
            const float msum = acc[j][0];
            const float fsum = acc[j][1];
            if (FUSED) {
                const float mean = (length != 1.0f) ? (msum / (float)count)
                                                    : (fsum * (1.0f / (float)T_));
                const float gmin = mm[0], gmax = mm[1];
                const float gg   = (graw[(size_t)b * D_ + d] - gmin) / (gmax - gmin);
                outp[(size_t)b * D_ + d] = mean + EPS_ * (-8.0f * gg + 9.0f);
            } else {
                const size_t idx = ((size_t)tc * B_ + b) * D_ + d;
                wsm[idx] = msum;
                wsf[idx] = fsum;
            }
        }
    }
}

// ---------------------------------------------------------------------------
// Kernel 3: fixed-order chunk reduction + mean select + noise (deterministic).
// ---------------------------------------------------------------------------
__global__ __launch_bounds__(256) void finalize_kernel(const float* __restrict__ lens,
                                                       const float* __restrict__ graw,
                                                       const float* __restrict__ mm,
                                                       const float* __restrict__ wsm,
                                                       const float* __restrict__ wsf,
                                                       float* __restrict__ outp,
                                                       int tchunks) {
    const int i = blockIdx.x * 256 + threadIdx.x;   // 0 .. B*D-1
    const int b = i / D_;
    const float length = lens[b] * (float)T_;
    int count = (int)floorf(length) + 1;
    count = count < 1 ? 1 : (count > T_ ? T_ : count);

    float msum = 0.f, fsum = 0.f;
    for (int c = 0; c < tchunks; ++c) {
        msum += wsm[(size_t)c * B_ * D_ + i];
        fsum += wsf[(size_t)c * B_ * D_ + i];
    }
    const float mean = (length != 1.0f) ? (msum / (float)count)
                                        : (fsum * (1.0f / (float)T_));
    const float gmin = mm[0], gmax = mm[1];
    const float gg   = (graw[i] - gmin) / (gmax - gmin);
    outp[i] = mean + EPS_ * (-8.0f * gg + 9.0f);
}

// ---------------------------------------------------------------------------
extern "C" void kernel_launch(void* const* d_in, const int* in_sizes, int n_in,
                              void* d_out, int out_size, void* d_ws, size_t ws_size,
                              hipStream_t stream) {
    (void)in_sizes; (void)n_in; (void)out_size;
    const float* xs   = (const float*)d_in[0];   // outputs [B,T,D] f32
    const float* lens = (const float*)d_in[1];   // lens    [B]     f32
    const float* graw = (const float*)d_in[2];   // gnoise  [B,D]   f32
    float*       outp = (float*)d_out;           // [B,1,D] f32

    // Choose t-split so partial sums fit in the workspace (2 arrays of
    // tchunks*B*D floats after a 256 B header for min/max).
    int tchunks = 8;
    while (tchunks > 1 &&
           (size_t)2 * (size_t)tchunks * B_ * D_ * sizeof(float) + 256 > ws_size)
        tchunks >>= 1;
    const bool fused =
        ((size_t)2 * (size_t)tchunks * B_ * D_ * sizeof(float) + 256 > ws_size);

    float* mm  = (float*)d_ws;                          // [0]=min, [1]=max
    float* wsm = (float*)((char*)d_ws + 256);           // masked partials
    float* wsf = wsm + (size_t)tchunks * B_ * D_;       // full partials

    gminmax_kernel<<<1, 1024, 0, stream>>>(graw, mm);

    if (fused) {
        pool_kernel<true><<<dim3(1, B_), 256, 0, stream>>>(
            xs, lens, graw, mm, nullptr, nullptr, outp, 1);
    } else {
        pool_kernel<false><<<dim3(tchunks, B_), 256, 0, stream>>>(
            xs, lens, graw, mm, wsm, wsf, nullptr, tchunks);
        finalize_kernel<<<(B_ * D_) / 256, 256, 0, stream>>>(
            lens, graw, mm, wsm, wsf, outp, tchunks);
    }
}